// StateEncoder6QDummy_86182813762082
// MI455X (gfx1250) — compile-verified
//
#include <hip/hip_runtime.h>
#include <hip/hip_bf16.h>

// ---------------------------------------------------------------------------
// StateEncoder6Q: theta = colsum(x[B,4]); state = kron(RX(theta_i)|0>, |0>,|0>)
// out[B,64] complex64 = broadcast(state).
//
// Stage 1: column-sum reduction via V_WMMA_F32_16X16X4_F32 (full f32 WMMA).
//          Batch on K (4 rows/WMMA) x 4 batch-groups packed into N -> each
//          wmma consumes a 16-row tile. Per-wave partials -> workspace.
// Stage 2: single block folds partials in a FIXED order (deterministic, no
//          float atomics), builds the 64-complex state (128 floats) in ws.
// Stage 3: bandwidth-bound broadcast: each thread owns one float4 of the
//          state in registers and streams global_store_b128 to d_out.
// ---------------------------------------------------------------------------

typedef __attribute__((ext_vector_type(2))) float v2f;
typedef __attribute__((ext_vector_type(8))) float v8f;

#define S1_BLOCKS 64
#define S1_THREADS 256
#define S1_WAVES ((S1_BLOCKS * S1_THREADS) / 32) // 512 waves

#define S3_BLOCKS 1024
#define S3_THREADS 256

// ---------------- Stage 1: WMMA column-sum reduction -----------------------
// partials layout: [S1_WAVES][16] floats; partials[w*16+n] = sum over this
// wave's tiles of x[4*(n>>2)+k + 16t, n&3] summed over k (column n&3).
__global__ void enc6q_reduce_wmma(const float* __restrict__ x,
                                  float* __restrict__ partials,
                                  int totalTiles /* B/16 */) {
  const int lane = threadIdx.x & 31;
  const int wave = (blockIdx.x * blockDim.x + threadIdx.x) >> 5;

  const int n     = lane & 15;
  const int khalf = lane >> 4;
  // B-matrix element for VGPR0 of this lane: row = (n&12) + 2*khalf, col = n&3
  const int idx0 = (n & 12) * 4 + khalf * 8 + (n & 3);

  v2f a;            // A = ones(16x4): every lane, both K slots = 1.0
  a.x = 1.0f;
  a.y = 1.0f;
  v8f c = {};       // f32 accumulator (16x16)

  for (int t = wave; t < totalTiles; t += S1_WAVES) {
    const float* base = x + (long long)t * 64;  // 16 rows x 4 cols
    v2f b;
    b.x = base[idx0];       // K = 2*khalf
    b.y = base[idx0 + 4];   // K = 2*khalf + 1 (next row)
    // D = A*B + C : contracts over K (4 batch rows), accumulates in C.
    c = __builtin_amdgcn_wmma_f32_16x16x4_f32(
        /*neg_a=*/false, a, /*neg_b=*/false, b,
        /*c_mod=*/(short)0, c, /*reuse_a=*/false, /*reuse_b=*/false);
  }

  // Accumulator row M=0 lives in VGPR0 (c[0]) on lanes 0..15 (N = lane).
  if (lane < 16) {
    partials[wave * 16 + lane] = c[0];
  }
}

// ---------------- Stage 2: fold partials + build 64-state ------------------
// state: 128 floats (64 complex). Fixed-order summation => deterministic.
__global__ void enc6q_build_state(const float* __restrict__ partials,
                                  float* __restrict__ state) {
  const int t = threadIdx.x;  // 0..63, state amplitude index (wire0 outermost)

  // theta[j] = sum over all waves w, groups g of partials[w*16 + 4g + j]
  float th0 = 0.f, th1 = 0.f, th2 = 0.f, th3 = 0.f;
  for (int w = 0; w < S1_WAVES; ++w) {
    const float* p = partials + w * 16;
#pragma unroll
    for (int g = 0; g < 4; ++g) {
      th0 += p[4 * g + 0];
      th1 += p[4 * g + 1];
      th2 += p[4 * g + 2];
      th3 += p[4 * g + 3];
    }
  }

  float cs[4], sn[4];
  cs[0] = __cosf(0.5f * th0); sn[0] = __sinf(0.5f * th0);
  cs[1] = __cosf(0.5f * th1); sn[1] = __sinf(0.5f * th1);
  cs[2] = __cosf(0.5f * th2); sn[2] = __sinf(0.5f * th2);
  cs[3] = __cosf(0.5f * th3); sn[3] = __sinf(0.5f * th3);
  // full-accuracy fallback for large |theta| (sum of 262144 normals ~ +-1500):
  cs[0] = cosf(0.5f * th0); sn[0] = sinf(0.5f * th0);
  cs[1] = cosf(0.5f * th1); sn[1] = sinf(0.5f * th1);
  cs[2] = cosf(0.5f * th2); sn[2] = sinf(0.5f * th2);
  cs[3] = cosf(0.5f * th3); sn[3] = sinf(0.5f * th3);

  // amplitude(t) = prod_{i=0..3} (bit_i ? -i*sin(h_i) : cos(h_i)) * [b4==b5==0]
  // index bits: b0 = t>>5 (wire 0, outermost) ... b5 = t&1.
  float mag = 1.0f;
  int p = 0;
#pragma unroll
  for (int i = 0; i < 4; ++i) {
    const int bit = (t >> (5 - i)) & 1;
    mag *= bit ? sn[i] : cs[i];
    p += bit;
  }
  float re, im;
  switch (p & 3) {              // multiply by (-i)^p
    case 0:  re = mag;  im = 0.0f; break;
    case 1:  re = 0.0f; im = -mag; break;
    case 2:  re = -mag; im = 0.0f; break;
    default: re = 0.0f; im = mag;  break;
  }
  if ((t & 3) != 0) { re = 0.0f; im = 0.0f; }  // wires 4,5 stay |0>

  state[2 * t + 0] = re;
  state[2 * t + 1] = im;
}

// ---------------- Stage 3: streaming broadcast (store-bound) ---------------
// out viewed as float4[B*32]; out4[i] = state4[i & 31]. Grid stride is a
// multiple of 32, so each thread's state slice is loop-invariant -> held in
// registers, pure global_store_b128 stream at HBM rate.
__global__ void enc6q_broadcast(const float4* __restrict__ state4,
                                float4* __restrict__ out4,
                                long long total4) {
  const long long stride = (long long)gridDim.x * blockDim.x;   // 262144 (mult of 32)
  long long i = (long long)blockIdx.x * blockDim.x + threadIdx.x;
  const float4 v = state4[(int)(i & 31)];
  for (; i < total4; i += stride) {
    out4[i] = v;
  }
}

// ---------------------------------------------------------------------------
extern "C" void kernel_launch(void* const* d_in, const int* in_sizes, int n_in,
                              void* d_out, int out_size, void* d_ws, size_t ws_size,
                              hipStream_t stream) {
  const float* x = (const float*)d_in[0];
  const int B = in_sizes[0] / 4;          // rows of x (262144)
  const int totalTiles = B / 16;          // 16 rows per WMMA tile

  float* partials = (float*)d_ws;                         // S1_WAVES*16 floats (32 KB)
  float* state    = (float*)((char*)d_ws + S1_WAVES * 16 * sizeof(float));

  enc6q_reduce_wmma<<<S1_BLOCKS, S1_THREADS, 0, stream>>>(x, partials, totalTiles);
  enc6q_build_state<<<1, 64, 0, stream>>>(partials, state);

  const long long total4 = (long long)B * 32;             // B * 128 floats / 4
  enc6q_broadcast<<<S3_BLOCKS, S3_THREADS, 0, stream>>>(
      (const float4*)state, (float4*)d_out, total4);
}